// GraphMLP_80668075753591
// MI455X (gfx1250) — compile-verified
//
#include <hip/hip_runtime.h>
#include <math.h>

#define NODES_CH 64
#define HEADS 4
#define HC 16
#define NG 16
#define GDIM 24
#define LIN_DIM 32
#define FEATD 88
#define EPSV 1e-5f

typedef __attribute__((ext_vector_type(2))) float v2f;
typedef __attribute__((ext_vector_type(8))) float v8f;

// ---------------- utility device functions ----------------
__device__ __forceinline__ float gelu_exact(float x) {
    return 0.5f * x * (1.0f + erff(x * 0.70710678118654752f));
}

__device__ __forceinline__ float atomic_max_float(float* addr, float val) {
    int* ia = (int*)addr;
    int old = __float_as_int(*addr);
    while (__int_as_float(old) < val) {
        int assumed = old;
        old = atomicCAS(ia, assumed, __float_as_int(val));
        if (old == assumed) break;
    }
    return __int_as_float(old);
}

__device__ __forceinline__ float wave_sum32(float v) {
    for (int o = 16; o; o >>= 1) v += __shfl_xor(v, o, 32);
    return v;
}

// ---------------- generic fills ----------------
__global__ void gml_fill(float* __restrict__ p, int n, float v) {
    int i = blockIdx.x * blockDim.x + threadIdx.x;
    if (i < n) p[i] = v;
}

__global__ void gml_count_nodes(const int* __restrict__ batch, float* __restrict__ cnt, int N) {
    int i = blockIdx.x * blockDim.x + threadIdx.x;
    if (i < N) atomicAdd(&cnt[batch[i]], 1.0f);
}

// ---------------- dense linear via fp32 WMMA ----------------
// Specialized K=64: Y[N,64] = X[N,64] @ W[64,64]; fully unrolled 16 WMMA steps.
// One wave computes a 16x16 tile; blockDim (32,4) covers 16 rows x 64 cols.
__global__ void gml_linear_wmma64(const float* __restrict__ X, const float* __restrict__ W,
                                  float* __restrict__ Y, int N) {
    int lane = threadIdx.x;          // 0..31
    int nbase = threadIdx.y * 16;    // 0,16,32,48
    int mbase = blockIdx.x * 16;
    int hi = lane >> 4;              // 0 or 1
    int lo = lane & 15;
    int row = mbase + lo;
    int rowc = (row < N) ? row : (N - 1);     // clamp: loads always in-bounds
    const float* xrow = X + (size_t)rowc * 64;
    const float* wcol = W + nbase + lo;
    v8f c = {};
    #pragma unroll
    for (int k = 0; k < 64; k += 4) {
        int ka = k + 2 * hi;                  // even -> 8B-aligned pair
        v2f a = *(const v2f*)(xrow + ka);     // global_load_b64
        v2f b;
        b.x = wcol[(size_t)ka * 64];
        b.y = wcol[(size_t)(ka + 1) * 64];
        c = __builtin_amdgcn_wmma_f32_16x16x4_f32(false, a, false, b, (short)0, c,
                                                  false, false);
    }
    #pragma unroll
    for (int r = 0; r < 8; ++r) {
        int orow = mbase + r + 8 * hi;
        if (orow < N) Y[(size_t)orow * 64 + nbase + lo] = c[r];
    }
}

// Generic K (layer 0, K=261): branchless main loop + one masked tail step.
__global__ void gml_linear_wmma_gen(const float* __restrict__ X, const float* __restrict__ W,
                                    float* __restrict__ Y, int N, int K) {
    int lane = threadIdx.x;
    int nbase = threadIdx.y * 16;
    int mbase = blockIdx.x * 16;
    int hi = lane >> 4;
    int lo = lane & 15;
    int row = mbase + lo;
    int rowc = (row < N) ? row : (N - 1);
    const float* xrow = X + (size_t)rowc * K;
    const float* wcol = W + nbase + lo;
    v8f c = {};
    int kmain = K & ~3;
    #pragma unroll 4
    for (int k = 0; k < kmain; k += 4) {
        int ka = k + 2 * hi;
        v2f a, b;
        a.x = xrow[ka];
        a.y = xrow[ka + 1];
        b.x = wcol[(size_t)ka * 64];
        b.y = wcol[(size_t)(ka + 1) * 64];
        c = __builtin_amdgcn_wmma_f32_16x16x4_f32(false, a, false, b, (short)0, c,
                                                  false, false);
    }
    if (kmain < K) {
        int ka = kmain + 2 * hi;
        int k0 = (ka < K) ? ka : (K - 1);         // clamped, always valid addr
        int k1 = (ka + 1 < K) ? (ka + 1) : (K - 1);
        float ax = xrow[k0], ay = xrow[k1];
        float bx = wcol[(size_t)k0 * 64], by = wcol[(size_t)k1 * 64];
        v2f a, b;
        a.x = (ka     < K) ? ax : 0.0f;           // mask value, not the load
        a.y = (ka + 1 < K) ? ay : 0.0f;
        b.x = (ka     < K) ? bx : 0.0f;
        b.y = (ka + 1 < K) ? by : 0.0f;
        c = __builtin_amdgcn_wmma_f32_16x16x4_f32(false, a, false, b, (short)0, c,
                                                  false, false);
    }
    #pragma unroll
    for (int r = 0; r < 8; ++r) {
        int orow = mbase + r + 8 * hi;
        if (orow < N) Y[(size_t)orow * 64 + nbase + lo] = c[r];
    }
}

// ---------------- attention logits: es/ed [N,4] ----------------
__global__ void gml_attn_logits(const float* __restrict__ lin,
                                const float* __restrict__ asrc, const float* __restrict__ adst,
                                float* __restrict__ es, float* __restrict__ ed, int N) {
    int i = blockIdx.x * blockDim.x + threadIdx.x;   // N*4
    if (i >= N * 4) return;
    int n = i >> 2, h = i & 3;
    const float* hp = lin + (size_t)n * 64 + h * 16;
    float s = 0.f, d = 0.f;
    #pragma unroll
    for (int c = 0; c < 16; ++c) {
        float v = hp[c];
        s += v * asrc[h * 16 + c];
        d += v * adst[h * 16 + c];
    }
    es[i] = s;
    ed[i] = d;
}

__device__ __forceinline__ void edge_sd(const int* ei, int E, int e, int& s, int& d) {
    if (e < E) { s = ei[e]; d = ei[E + e]; }
    else       { s = d = e - E; }   // self loop
}

// pass 1: segment max of leaky-relu logits into emax[N,4]
__global__ void gml_edge_max(const int* __restrict__ ei, int E, int N,
                             const float* __restrict__ es, const float* __restrict__ ed,
                             float* __restrict__ emax) {
    int i = blockIdx.x * blockDim.x + threadIdx.x;
    int tot = (E + N) * 4;
    if (i >= tot) return;
    int e = i >> 2, h = i & 3, s, d;
    edge_sd(ei, E, e, s, d);
    float x = es[s * 4 + h] + ed[d * 4 + h];
    x = (x > 0.f) ? x : 0.2f * x;
    atomic_max_float(&emax[d * 4 + h], x);
}

// pass 2: denom[N,4] += exp(x - emax)
__global__ void gml_edge_den(const int* __restrict__ ei, int E, int N,
                             const float* __restrict__ es, const float* __restrict__ ed,
                             const float* __restrict__ emax, float* __restrict__ denom) {
    int i = blockIdx.x * blockDim.x + threadIdx.x;
    int tot = (E + N) * 4;
    if (i >= tot) return;
    int e = i >> 2, h = i & 3, s, d;
    edge_sd(ei, E, e, s, d);
    float x = es[s * 4 + h] + ed[d * 4 + h];
    x = (x > 0.f) ? x : 0.2f * x;
    atomicAdd(&denom[d * 4 + h], expf(x - emax[d * 4 + h]));
}

// pass 3: agg[d, h*16+c] += h[s, h*16+c] * alpha
__global__ void gml_edge_agg(const int* __restrict__ ei, int E, int N,
                             const float* __restrict__ es, const float* __restrict__ ed,
                             const float* __restrict__ emax, const float* __restrict__ denom,
                             const float* __restrict__ lin, float* __restrict__ agg) {
    int i = blockIdx.x * blockDim.x + threadIdx.x;
    int tot = (E + N) * 4;
    if (i >= tot) return;
    int e = i >> 2, h = i & 3, s, d;
    edge_sd(ei, E, e, s, d);
    float x = es[s * 4 + h] + ed[d * 4 + h];
    x = (x > 0.f) ? x : 0.2f * x;
    int di = d * 4 + h;
    float alpha = expf(x - emax[di]) / (denom[di] + 1e-16f);
    const float* hs = lin + (size_t)s * 64 + h * 16;
    float* ap = agg + (size_t)d * 64 + h * 16;
    #pragma unroll
    for (int c = 0; c < 16; ++c) atomicAdd(&ap[c], hs[c] * alpha);
}

// ---------------- GraphNorm passes ----------------
__global__ void gml_gelu_gsum(float* __restrict__ agg, const float* __restrict__ bias,
                              const int* __restrict__ batch, float* __restrict__ gsum, int N) {
    int i = blockIdx.x * blockDim.x + threadIdx.x;
    if (i >= N * 64) return;
    int n = i >> 6, c = i & 63;
    float v = gelu_exact(agg[i] + bias[c]);
    agg[i] = v;
    atomicAdd(&gsum[batch[n] * 64 + c], v);
}

__global__ void gml_mean(const float* __restrict__ gsum, const float* __restrict__ cnt,
                         float* __restrict__ mean) {
    int i = blockIdx.x * blockDim.x + threadIdx.x;   // 16*64
    if (i >= NG * 64) return;
    float c = cnt[i >> 6];
    mean[i] = gsum[i] / fmaxf(c, 1.0f);
}

__global__ void gml_var_gsum(const float* __restrict__ agg, const float* __restrict__ mean,
                             const float* __restrict__ ms, const int* __restrict__ batch,
                             float* __restrict__ gsum2, int N) {
    int i = blockIdx.x * blockDim.x + threadIdx.x;
    if (i >= N * 64) return;
    int n = i >> 6, c = i & 63, g = batch[n];
    float o = agg[i] - mean[g * 64 + c] * ms[c];
    atomicAdd(&gsum2[g * 64 + c], o * o);
}

__global__ void gml_invstd(const float* __restrict__ gsum2, const float* __restrict__ cnt,
                           float* __restrict__ invstd) {
    int i = blockIdx.x * blockDim.x + threadIdx.x;
    if (i >= NG * 64) return;
    float c = fmaxf(cnt[i >> 6], 1.0f);
    invstd[i] = rsqrtf(gsum2[i] / c + EPSV);
}

__global__ void gml_gn_final(const float* __restrict__ agg, const float* __restrict__ mean,
                             const float* __restrict__ invstd, const float* __restrict__ ms,
                             const float* __restrict__ w, const float* __restrict__ b,
                             const int* __restrict__ batch, float* __restrict__ cur,
                             int N, int residual) {
    int i = blockIdx.x * blockDim.x + threadIdx.x;
    if (i >= N * 64) return;
    int n = i >> 6, c = i & 63, g = batch[n];
    float o = agg[i] - mean[g * 64 + c] * ms[c];
    float y = w[c] * o * invstd[g * 64 + c] + b[c];
    cur[i] = residual ? (cur[i] + y) : y;
}

// ---------------- pooling + feature concat ----------------
__global__ void gml_pool_sum(const float* __restrict__ cur, const int* __restrict__ batch,
                             float* __restrict__ gsum, int N) {
    int i = blockIdx.x * blockDim.x + threadIdx.x;
    if (i >= N * 64) return;
    int n = i >> 6, c = i & 63;
    atomicAdd(&gsum[batch[n] * 64 + c], cur[i]);
}

__global__ void gml_build_feat(const float* __restrict__ gsum, const float* __restrict__ cnt,
                               const float* __restrict__ gf, float* __restrict__ feat) {
    int i = blockIdx.x * blockDim.x + threadIdx.x;   // 16*88
    if (i >= NG * FEATD) return;
    int g = i / FEATD, c = i % FEATD;
    feat[i] = (c < 64) ? gsum[g * 64 + c] / fmaxf(cnt[g], 1.0f)
                       : gf[g * GDIM + (c - 64)];
}

// ---------------- MLP head: 1 wave32 per graph ----------------
__global__ void gml_mlp_head(const float* __restrict__ feat,
                             const float* __restrict__ W0, const float* __restrict__ b0,
                             const float* __restrict__ lnw0, const float* __restrict__ lnb0,
                             const float* __restrict__ Wg, const float* __restrict__ bg,
                             const float* __restrict__ lnw, const float* __restrict__ lnb,
                             const float* __restrict__ hW, const float* __restrict__ hb,
                             float* __restrict__ out) {
    int g = blockIdx.x;      // 0..15
    int j = threadIdx.x;     // 0..31
    __shared__ float fr[FEATD];
    __shared__ float zs[LIN_DIM];
    for (int c = j; c < FEATD; c += 32) fr[c] = feat[g * FEATD + c];
    __syncthreads();

    // block 0: 88 -> 32, gelu, LN
    float a = b0[j];
    for (int c = 0; c < FEATD; ++c) a += fr[c] * W0[c * LIN_DIM + j];
    a = gelu_exact(a);
    float m = wave_sum32(a) * (1.0f / 32.0f);
    float dd = a - m;
    float var = wave_sum32(dd * dd) * (1.0f / 32.0f);
    float z = dd * rsqrtf(var + EPSV) * lnw0[j] + lnb0[j];

    // residual blocks
    for (int l = 0; l < 3; ++l) {
        zs[j] = z;
        __syncthreads();
        float acc = bg[l * LIN_DIM + j];
        for (int c = 0; c < LIN_DIM; ++c)
            acc += zs[c] * Wg[l * LIN_DIM * LIN_DIM + c * LIN_DIM + j];
        __syncthreads();
        acc = gelu_exact(acc);
        float mm = wave_sum32(acc) * (1.0f / 32.0f);
        float d2 = acc - mm;
        float v2 = wave_sum32(d2 * d2) * (1.0f / 32.0f);
        z = d2 * rsqrtf(v2 + EPSV) * lnw[l * LIN_DIM + j] + lnb[l * LIN_DIM + j] + z;
    }

    float p = wave_sum32(z * hW[j]);
    if (j == 0) out[g] = p + hb[0];
}

// ---------------- launcher ----------------
static inline int gb(int n, int b) { return (n + b - 1) / b; }

extern "C" void kernel_launch(void* const* d_in, const int* in_sizes, int n_in,
                              void* d_out, int out_size, void* d_ws, size_t ws_size,
                              hipStream_t stream) {
    const float* x      = (const float*)d_in[0];
    const int*   ei     = (const int*)d_in[1];
    const int*   batch  = (const int*)d_in[2];
    const float* gf     = (const float*)d_in[3];
    const float* W0     = (const float*)d_in[4];
    const float* asrc0  = (const float*)d_in[5];
    const float* adst0  = (const float*)d_in[6];
    const float* b0     = (const float*)d_in[7];
    const float* gnw0   = (const float*)d_in[8];
    const float* gnb0   = (const float*)d_in[9];
    const float* gnms0  = (const float*)d_in[10];
    const float* Wg     = (const float*)d_in[11];
    const float* asrcg  = (const float*)d_in[12];
    const float* adstg  = (const float*)d_in[13];
    const float* bg     = (const float*)d_in[14];
    const float* gnwg   = (const float*)d_in[15];
    const float* gnbg   = (const float*)d_in[16];
    const float* gnmsg  = (const float*)d_in[17];
    const float* mW0    = (const float*)d_in[18];
    const float* mb0    = (const float*)d_in[19];
    const float* lnw0   = (const float*)d_in[20];
    const float* lnb0   = (const float*)d_in[21];
    const float* mWg    = (const float*)d_in[22];
    const float* mbg    = (const float*)d_in[23];
    const float* lnwg   = (const float*)d_in[24];
    const float* lnbg   = (const float*)d_in[25];
    const float* hW     = (const float*)d_in[26];
    const float* hb     = (const float*)d_in[27];

    const int N = in_sizes[2];          // batch length = number of nodes
    const int E = in_sizes[1] / 2;      // edge_index is [2,E]
    const int K0 = in_sizes[0] / N;     // 261

    float* ws = (float*)d_ws;
    size_t o = 0;
    float* cur    = ws + o; o += (size_t)N * 64;
    float* lin    = ws + o; o += (size_t)N * 64;
    float* agg    = ws + o; o += (size_t)N * 64;
    float* es     = ws + o; o += (size_t)N * 4;
    float* ed     = ws + o; o += (size_t)N * 4;
    float* emax   = ws + o; o += (size_t)N * 4;
    float* denom  = ws + o; o += (size_t)N * 4;
    float* cnt    = ws + o; o += NG;
    float* gsum   = ws + o; o += NG * 64;
    float* gsum2  = ws + o; o += NG * 64;
    float* mean   = ws + o; o += NG * 64;
    float* invstd = ws + o; o += NG * 64;
    float* feat   = ws + o; o += NG * FEATD;

    const int B = 256;
    const int NC = N * 64;
    const int NH = N * 4;
    const int EH = (E + N) * 4;

    // per-graph node counts (recomputed every call: deterministic)
    gml_fill<<<1, 64, 0, stream>>>(cnt, NG, 0.0f);
    gml_count_nodes<<<gb(N, B), B, 0, stream>>>(batch, cnt, N);

    for (int l = 0; l < 7; ++l) {
        const float* Xin  = (l == 0) ? x : cur;
        const float* W    = (l == 0) ? W0 : (Wg + (size_t)(l - 1) * 64 * 64);
        const float* as   = (l == 0) ? asrc0 : (asrcg + (size_t)(l - 1) * HEADS * HC);
        const float* ad   = (l == 0) ? adst0 : (adstg + (size_t)(l - 1) * HEADS * HC);
        const float* bias = (l == 0) ? b0 : (bg + (size_t)(l - 1) * 64);
        const float* gw   = (l == 0) ? gnw0 : (gnwg + (size_t)(l - 1) * 64);
        const float* gbb  = (l == 0) ? gnb0 : (gnbg + (size_t)(l - 1) * 64);
        const float* gms  = (l == 0) ? gnms0 : (gnmsg + (size_t)(l - 1) * 64);

        // lin = Xin @ W  (fp32 WMMA)
        if (l == 0)
            gml_linear_wmma_gen<<<dim3(gb(N, 16)), dim3(32, 4), 0, stream>>>(Xin, W, lin, N, K0);
        else
            gml_linear_wmma64<<<dim3(gb(N, 16)), dim3(32, 4), 0, stream>>>(Xin, W, lin, N);

        // attention
        gml_attn_logits<<<gb(NH, B), B, 0, stream>>>(lin, as, ad, es, ed, N);
        gml_fill<<<gb(NH, B), B, 0, stream>>>(emax, NH, -3.0e38f);
        gml_fill<<<gb(NH, B), B, 0, stream>>>(denom, NH, 0.0f);
        gml_fill<<<gb(NC, B), B, 0, stream>>>(agg, NC, 0.0f);
        gml_edge_max<<<gb(EH, B), B, 0, stream>>>(ei, E, N, es, ed, emax);
        gml_edge_den<<<gb(EH, B), B, 0, stream>>>(ei, E, N, es, ed, emax, denom);
        gml_edge_agg<<<gb(EH, B), B, 0, stream>>>(ei, E, N, es, ed, emax, denom, lin, agg);

        // gelu + GraphNorm (+ residual for l>0)
        gml_fill<<<gb(NG * 64, B), B, 0, stream>>>(gsum, NG * 64, 0.0f);
        gml_gelu_gsum<<<gb(NC, B), B, 0, stream>>>(agg, bias, batch, gsum, N);
        gml_mean<<<gb(NG * 64, B), B, 0, stream>>>(gsum, cnt, mean);
        gml_fill<<<gb(NG * 64, B), B, 0, stream>>>(gsum2, NG * 64, 0.0f);
        gml_var_gsum<<<gb(NC, B), B, 0, stream>>>(agg, mean, gms, batch, gsum2, N);
        gml_invstd<<<gb(NG * 64, B), B, 0, stream>>>(gsum2, cnt, invstd);
        gml_gn_final<<<gb(NC, B), B, 0, stream>>>(agg, mean, invstd, gms, gw, gbb,
                                                  batch, cur, N, (l > 0) ? 1 : 0);
    }

    // global mean pool + concat globals
    gml_fill<<<gb(NG * 64, B), B, 0, stream>>>(gsum, NG * 64, 0.0f);
    gml_pool_sum<<<gb(NC, B), B, 0, stream>>>(cur, batch, gsum, N);
    gml_build_feat<<<gb(NG * FEATD, B), B, 0, stream>>>(gsum, cnt, gf, feat);

    // MLP head -> out[16]
    gml_mlp_head<<<NG, 32, 0, stream>>>(feat, mW0, mb0, lnw0, lnb0,
                                        mWg, mbg, lnwg, lnbg, hW, hb, (float*)d_out);
}